// SympNet_7121055777466
// MI455X (gfx1250) — compile-verified
//
#include <hip/hip_runtime.h>
#include <math.h>

#define NPTS    2097152
#define NLAYERS 32
#define NSUB    8
#define KPT     16     // points per thread (8 x v2f vectors)
#define NV      (KPT/2)
#define TPB     256    // 8 wave32 waves per block

typedef __attribute__((ext_vector_type(2))) float v2f;

// CDNA5 hardware tanh (single v_tanh_f32 TRANS op, co-executes with VALU).
__device__ __forceinline__ float htanh(float x) {
#if defined(__has_builtin)
#if __has_builtin(__builtin_amdgcn_tanhf)
  return __builtin_amdgcn_tanhf(x);
#else
  return tanhf(x);
#endif
#else
  return tanhf(x);
#endif
}

__device__ __forceinline__ v2f vfma(v2f a, v2f b, v2f c) {
  return __builtin_elementwise_fma(a, b, c);
}
__device__ __forceinline__ v2f bc(float s) { v2f r; r.x = s; r.y = s; return r; }

// Per layer i, compose the 8 shears into M = S7*...*S0 (det 1) and
// precompute c = M * b so the forward layer is a pure FMA chain.
// Record layout (8 floats): m00 m01 m10 m11 c0 c1 b0 b1
__global__ void symp_compose_kernel(const float* __restrict__ lin_w,
                                    const float2* __restrict__ bias_b,
                                    float* __restrict__ W) {
  int i = threadIdx.x;
  if (i >= NLAYERS) return;
  float m00 = 1.f, m01 = 0.f, m10 = 0.f, m11 = 1.f;
#pragma unroll
  for (int j = 0; j < NSUB; ++j) {
    float w = lin_w[i * NSUB + j];
    if ((j & 1) == 0) {            // q += w*p  -> row0 += w*row1
      m00 = fmaf(w, m10, m00);
      m01 = fmaf(w, m11, m01);
    } else {                       // p += w*q  -> row1 += w*row0
      m10 = fmaf(w, m00, m10);
      m11 = fmaf(w, m01, m11);
    }
  }
  float2 b = bias_b[i];
  float c0 = fmaf(m00, b.x, m01 * b.y);
  float c1 = fmaf(m10, b.x, m11 * b.y);
  float* r = W + i * 8;
  r[0] = m00; r[1] = m01; r[2] = m10; r[3] = m11;
  r[4] = c0;  r[5] = c1;  r[6] = b.x; r[7] = b.y;
}

__global__ __launch_bounds__(TPB) void symp_chain_kernel(
    const float2* __restrict__ x,
    const float*  __restrict__ act_w,
    const float4* __restrict__ W,     // 2 x float4 per layer
    float2*       __restrict__ out) {
  const int base = blockIdx.x * (TPB * KPT) + threadIdx.x;

  v2f qv[NV], pv[NV];
#pragma unroll
  for (int j = 0; j < NV; ++j) {
    float2 a = x[base + (2 * j + 0) * TPB];   // coalesced global_load_b64
    float2 b = x[base + (2 * j + 1) * TPB];
    qv[j].x = a.x; qv[j].y = b.x;
    pv[j].x = a.y; pv[j].y = b.y;
  }

  // ---------------- forward chain (branch-free: even+odd layer pair) -------
#pragma unroll 1
  for (int i = 0; i < NLAYERS; i += 2) {
    {   // even layer: q += aw*tanh(p), then (q,p) <- M*(q,p) + M*b
      const v2f aw2 = bc(act_w[i]);
      const float4 A = W[2 * i], B = W[2 * i + 1];
      const v2f m00 = bc(A.x), m01 = bc(A.y), m10 = bc(A.z), m11 = bc(A.w);
      const v2f c0 = bc(B.x), c1 = bc(B.y);
#pragma unroll
      for (int j = 0; j < NV; ++j) {
        v2f t; t.x = htanh(pv[j].x); t.y = htanh(pv[j].y);
        qv[j] = vfma(aw2, t, qv[j]);
      }
#pragma unroll
      for (int j = 0; j < NV; ++j) {
        v2f qn = vfma(m00, qv[j], vfma(m01, pv[j], c0));
        v2f pn = vfma(m10, qv[j], vfma(m11, pv[j], c1));
        qv[j] = qn; pv[j] = pn;
      }
    }
    {   // odd layer: p += aw*tanh(q), then matvec
      const v2f aw2 = bc(act_w[i + 1]);
      const float4 A = W[2 * i + 2], B = W[2 * i + 3];
      const v2f m00 = bc(A.x), m01 = bc(A.y), m10 = bc(A.z), m11 = bc(A.w);
      const v2f c0 = bc(B.x), c1 = bc(B.y);
#pragma unroll
      for (int j = 0; j < NV; ++j) {
        v2f t; t.x = htanh(qv[j].x); t.y = htanh(qv[j].y);
        pv[j] = vfma(aw2, t, pv[j]);
      }
#pragma unroll
      for (int j = 0; j < NV; ++j) {
        v2f qn = vfma(m00, qv[j], vfma(m01, pv[j], c0));
        v2f pn = vfma(m10, qv[j], vfma(m11, pv[j], c1));
        qv[j] = qn; pv[j] = pn;
      }
    }
  }

  // ---------------- time reversal ----------------
#pragma unroll
  for (int j = 0; j < NV; ++j) pv[j] = -pv[j];

  // ---------------- inverse chain (odd layer then even layer per pair) -----
  // inverse matrix is the adjugate (det(M)==1): [[m11,-m01],[-m10,m00]];
  // bias removal folds into the innermost FMA addend (-b).
#pragma unroll 1
  for (int i = NLAYERS - 2; i >= 0; i -= 2) {
    {   // odd layer i+1: adjugate matvec - b, then p -= aw*tanh(q)
      const v2f naw2 = bc(-act_w[i + 1]);
      const float4 A = W[2 * i + 2], B = W[2 * i + 3];
      const v2f m00 = bc(A.x), nm01 = bc(-A.y), nm10 = bc(-A.z), m11 = bc(A.w);
      const v2f nb0 = bc(-B.z), nb1 = bc(-B.w);
#pragma unroll
      for (int j = 0; j < NV; ++j) {
        v2f qn = vfma(m11, qv[j], vfma(nm01, pv[j], nb0)); //  m11 q - m01 p - b0
        v2f pn = vfma(m00, pv[j], vfma(nm10, qv[j], nb1)); // -m10 q + m00 p - b1
        qv[j] = qn; pv[j] = pn;
      }
#pragma unroll
      for (int j = 0; j < NV; ++j) {
        v2f t; t.x = htanh(qv[j].x); t.y = htanh(qv[j].y);
        pv[j] = vfma(naw2, t, pv[j]);
      }
    }
    {   // even layer i: adjugate matvec - b, then q -= aw*tanh(p)
      const v2f naw2 = bc(-act_w[i]);
      const float4 A = W[2 * i], B = W[2 * i + 1];
      const v2f m00 = bc(A.x), nm01 = bc(-A.y), nm10 = bc(-A.z), m11 = bc(A.w);
      const v2f nb0 = bc(-B.z), nb1 = bc(-B.w);
#pragma unroll
      for (int j = 0; j < NV; ++j) {
        v2f qn = vfma(m11, qv[j], vfma(nm01, pv[j], nb0));
        v2f pn = vfma(m00, pv[j], vfma(nm10, qv[j], nb1));
        qv[j] = qn; pv[j] = pn;
      }
#pragma unroll
      for (int j = 0; j < NV; ++j) {
        v2f t; t.x = htanh(pv[j].x); t.y = htanh(pv[j].y);
        qv[j] = vfma(naw2, t, qv[j]);
      }
    }
  }

  // final scale by [1, -1]
#pragma unroll
  for (int j = 0; j < NV; ++j) {
    out[base + (2 * j + 0) * TPB] = make_float2(qv[j].x, -pv[j].x);
    out[base + (2 * j + 1) * TPB] = make_float2(qv[j].y, -pv[j].y);
  }
}

extern "C" void kernel_launch(void* const* d_in, const int* in_sizes, int n_in,
                              void* d_out, int out_size, void* d_ws, size_t ws_size,
                              hipStream_t stream) {
  (void)in_sizes; (void)n_in; (void)out_size; (void)ws_size;
  const float2* x      = (const float2*)d_in[0];  // (N, 2) float32
  const float*  act_w  = (const float*)d_in[1];   // (32,)
  const float2* bias_b = (const float2*)d_in[2];  // (32, 2)
  const float*  lin_w  = (const float*)d_in[3];   // (32, 8)
  float*        Wws    = (float*)d_ws;            // 32 * 8 floats of scratch
  float2*       out    = (float2*)d_out;

  symp_compose_kernel<<<1, NLAYERS, 0, stream>>>(lin_w, bias_b, Wws);

  const int points_per_block = TPB * KPT;                  // 4096
  const int grid = NPTS / points_per_block;                // 512
  symp_chain_kernel<<<grid, TPB, 0, stream>>>(x, act_w, (const float4*)Wws, out);
}